// ModulatedDeformConv_44702019617361
// MI455X (gfx1250) — compile-verified
//
#include <hip/hip_runtime.h>

typedef __attribute__((ext_vector_type(2))) float v2f;
typedef __attribute__((ext_vector_type(4))) float v4f;
typedef __attribute__((ext_vector_type(8))) float v8f;

#define DGN   8
#define KWIN  9            // 3x3
#define C_IN  128
#define CG    16           // C_IN / DGN
#define COUT  128
#define HH    128
#define WW    128
#define KTOT  1152         // C_IN * KWIN  (GEMM reduction dim)
#define TILE_W 32          // 32 pixels = two 16-wide M tiles per block
#define WTILES (WW / TILE_W)

// One block = one (b, h, 32-pixel) tile.
// Phase 1: 256 threads sample the 1152x32 A-panel into LDS, interleaved so the
//          WMMA loop fetches both M-tiles' A fragments with ONE ds_load_b128.
// Phase 2: 8 waves; each wave owns a 16-wide Cout chunk and computes two
//          16(pixel) x 16(Cout) tiles; every B fragment feeds two chained
//          V_WMMA_F32_16X16X4_F32 ops.
// LDS layout (dwords): addr = (kk/4)*128 + pxl*8 + ((kk>>1)&1)*4 + tile*2 + (kk&1)
//   -> per lane (pxl = lane&15, pair = lane>>4) a b128 read at
//      (kk/4)*128 + pxl*8 + pair*4 yields {A0[kk+sel], A0[kk+sel+1],
//                                          A1[kk+sel], A1[kk+sel+1]}.
__global__ __launch_bounds__(256, 2)
void deform_conv_wmma(const float* __restrict__ x,
                      const float* __restrict__ offset,
                      const float* __restrict__ mask,
                      const float* __restrict__ weight,
                      const float* __restrict__ bias,
                      float* __restrict__ out)
{
    extern __shared__ float smem[];          // KTOT*32 floats = 147456 B
    const int tid  = threadIdx.x;
    const int lane = tid & 31;
    const int wid  = tid >> 5;

    const int id  = blockIdx.x;
    const int b   = id / (HH * WTILES);
    const int rem = id % (HH * WTILES);
    const int h   = rem / WTILES;
    const int w0  = (rem % WTILES) * TILE_W;

    // Prefetch this wave's weight panel (16 rows x 1152 f32 = 72KB) so it is
    // cache-hot by the time the WMMA loop starts.
    {
        const int n = (wid << 4) + (lane & 15);
        const float* wrow = weight + (size_t)n * KTOT;
        #pragma unroll
        for (int j = (lane >> 4); j < (KTOT / 64); j += 2)
            __builtin_prefetch(wrow + j * 64, 0, 3);
    }

    // ---------------- Phase 1: modulated deformable bilinear sampling ----------------
    // sample-point sp -> px = sp&31, t = sp>>5, k = t%9, dg = t/9
    for (int sp = tid; sp < DGN * KWIN * TILE_W; sp += 256) {
        const int px = sp & 31;
        const int t  = sp >> 5;
        const int k  = t % KWIN;
        const int dg = t / KWIN;
        const int w  = w0 + px;

        const int ky  = k / 3, kx = k % 3;
        const int och = dg * KWIN + k;
        const int hw  = h * WW + w;
        const float dy = offset[((size_t)(b * (2 * DGN * KWIN) + 2 * och + 0) * (HH * WW)) + hw];
        const float dx = offset[((size_t)(b * (2 * DGN * KWIN) + 2 * och + 1) * (HH * WW)) + hw];
        const float m  = mask  [((size_t)(b * (DGN * KWIN) + och)            * (HH * WW)) + hw];

        const float sy  = dy + (float)(h - 1 + ky);
        const float sx  = dx + (float)(w - 1 + kx);
        const float y0f = floorf(sy);
        const float x0f = floorf(sx);
        const float ly  = sy - y0f;
        const float lx  = sx - x0f;
        const int y0 = (int)y0f, x0 = (int)x0f;
        const int y1 = y0 + 1,  x1 = x0 + 1;

        const bool vy0 = (y0 >= 0) & (y0 < HH);
        const bool vy1 = (y1 >= 0) & (y1 < HH);
        const bool vx0 = (x0 >= 0) & (x0 < WW);
        const bool vx1 = (x1 >= 0) & (x1 < WW);

        // fold mask + validity into the 4 corner weights
        const float w00 = (vy0 & vx0) ? (1.f - ly) * (1.f - lx) * m : 0.f;
        const float w01 = (vy0 & vx1) ? (1.f - ly) * lx         * m : 0.f;
        const float w10 = (vy1 & vx0) ? ly         * (1.f - lx) * m : 0.f;
        const float w11 = (vy1 & vx1) ? ly         * lx         * m : 0.f;

        const int cy0 = min(max(y0, 0), HH - 1);
        const int cy1 = min(max(y1, 0), HH - 1);
        const int cx0 = min(max(x0, 0), WW - 1);
        const int cx1 = min(max(x1, 0), WW - 1);
        const int i00 = cy0 * WW + cx0;
        const int i01 = cy0 * WW + cx1;
        const int i10 = cy1 * WW + cx0;
        const int i11 = cy1 * WW + cx1;

        const int tile = px >> 4;
        const int pxl  = px & 15;
        // constant part of the interleaved LDS index for this sample point
        const int sbase = (pxl << 3) + (tile << 1);

        const float* xg = x + (size_t)(b * C_IN + dg * CG) * (HH * WW);
        int kk = (dg * CG) * KWIN + k;                 // reduction index of (c_full, k)
        #pragma unroll 4
        for (int c = 0; c < CG; ++c) {
            const float* xc = xg + (size_t)c * (HH * WW);
            float v = w00 * xc[i00];
            v = fmaf(w01, xc[i01], v);
            v = fmaf(w10, xc[i10], v);
            v = fmaf(w11, xc[i11], v);
            smem[((kk >> 2) << 7) + sbase + (((kk >> 1) & 1) << 2) + (kk & 1)] = v;
            kk += KWIN;
        }
    }
    __syncthreads();

    // ------------- Phase 2: 32x16x1152 GEMM via V_WMMA_F32_16X16X4_F32 -------------
    const int n0  = wid << 4;
    const int nl  = n0 + (lane & 15);            // this lane's Cout column (N)
    const int sel = (lane >> 4) << 1;            // K sub-pair: {0,1} or {2,3}
    const float* wrow = weight + (size_t)nl * KTOT;
    const float* aptr = smem + ((lane & 15) << 3) + ((lane >> 4) << 2);

    v8f acc0 = {};
    v8f acc1 = {};
    #pragma unroll 4
    for (int kk = 0; kk < KTOT; kk += 4) {
        v2f bf = *(const v2f*)(wrow + kk + sel);    // B: K x Cout (cache-hot)
        v4f q  = *(const v4f*)(aptr + (kk << 5));   // A tiles 0 and 1, one b128
        v2f a0 = { q[0], q[1] };
        v2f a1 = { q[2], q[3] };
        acc0 = __builtin_amdgcn_wmma_f32_16x16x4_f32(false, a0, false, bf,
                                                     (short)0, acc0, false, false);
        acc1 = __builtin_amdgcn_wmma_f32_16x16x4_f32(false, a1, false, bf,
                                                     (short)0, acc1, false, false);
    }

    // D layout: lane -> N (Cout), VGPR v -> M (pixel); each lane's 8 results per
    // tile are 8 consecutive w positions -> coalesced b128 stores.
    const float bv    = bias[nl];
    const int   wbase = w0 + ((lane >> 4) << 3);
    float* op = out + (((size_t)(b * COUT + nl) * HH) + h) * WW + wbase;
    float4 q;
    q = (float4){ acc0[0] + bv, acc0[1] + bv, acc0[2] + bv, acc0[3] + bv };
    *(float4*)(op + 0)  = q;
    q = (float4){ acc0[4] + bv, acc0[5] + bv, acc0[6] + bv, acc0[7] + bv };
    *(float4*)(op + 4)  = q;
    q = (float4){ acc1[0] + bv, acc1[1] + bv, acc1[2] + bv, acc1[3] + bv };
    *(float4*)(op + 16) = q;
    q = (float4){ acc1[4] + bv, acc1[5] + bv, acc1[6] + bv, acc1[7] + bv };
    *(float4*)(op + 20) = q;
}

extern "C" void kernel_launch(void* const* d_in, const int* in_sizes, int n_in,
                              void* d_out, int out_size, void* d_ws, size_t ws_size,
                              hipStream_t stream) {
    const float* x      = (const float*)d_in[0];
    const float* offset = (const float*)d_in[1];
    const float* mask   = (const float*)d_in[2];
    const float* weight = (const float*)d_in[3];
    const float* bias   = (const float*)d_in[4];
    float* out = (float*)d_out;

    const size_t lds = (size_t)KTOT * TILE_W * sizeof(float);   // 147456 B dynamic LDS
    (void)hipFuncSetAttribute((const void*)deform_conv_wmma,
                              hipFuncAttributeMaxDynamicSharedMemorySize, (int)lds);

    dim3 grid(4 * HH * WTILES);     // 2048 tiles of 32 pixels
    dim3 block(256);
    hipLaunchKernelGGL(deform_conv_wmma, grid, block, lds, stream,
                       x, offset, mask, weight, bias, out);
}